// simple_RGGC_regression_59502476919377
// MI455X (gfx1250) — compile-verified
//
#include <hip/hip_runtime.h>

#define N_NODES 100000
#define N_EDGES 1000000
#define HID 64
#define N_GRAPHS 64
#define EPSV 1e-5f

typedef __attribute__((ext_vector_type(2))) float v2f;
typedef __attribute__((ext_vector_type(8))) float v8f;

__device__ __forceinline__ float sigmoidf(float x) { return 1.0f / (1.0f + __expf(-x)); }

__global__ void zero_kernel(float* p, int n) {
    int i = blockIdx.x * blockDim.x + threadIdx.x;
    if (i < n) p[i] = 0.0f;
}

// Layer 1 linears (din = 1): k/q/v and agg initialized with skip(x).
__global__ void lin1_kernel(const float* __restrict__ X,
                            const float* __restrict__ Wk, const float* __restrict__ bk,
                            const float* __restrict__ Wq, const float* __restrict__ bq,
                            const float* __restrict__ Wv, const float* __restrict__ bv,
                            const float* __restrict__ Ws, const float* __restrict__ bs,
                            float* __restrict__ K, float* __restrict__ Q,
                            float* __restrict__ V, float* __restrict__ AGG) {
    int gid = blockIdx.x * blockDim.x + threadIdx.x;
    if (gid >= N_NODES * HID) return;
    int c = gid & 63;
    float x = X[gid >> 6];
    K[gid]   = x * Wk[c] + bk[c];
    Q[gid]   = x * Wq[c] + bq[c];
    V[gid]   = x * Wv[c] + bv[c];
    AGG[gid] = x * Ws[c] + bs[c];   // skip(x) seeds the aggregation buffer
}

// Layers 2-5: four fp32 64x64 linears via V_WMMA_F32_16X16X4_F32.
// Block = 32x8 (8 waves); each wave owns one (matrix, 16-col tile) combo.
// A (16x4 f32): lanes 0-15 carry M=0..15 / K={k0,k0+1}; lanes 16-31 carry K={k0+2,k0+3}.
// B (4x16 f32): B[k][n] = W[n][k]; same K-half split across lane halves.
// D (16x16 f32): VGPR i -> M=i (lanes 0-15), M=i+8 (lanes 16-31), N = lane&15.
__global__ __launch_bounds__(256) void lin_wmma_kernel(const float* __restrict__ in,
        const float* __restrict__ W0, const float* __restrict__ B0,
        const float* __restrict__ W1, const float* __restrict__ B1,
        const float* __restrict__ W2, const float* __restrict__ B2,
        const float* __restrict__ W3, const float* __restrict__ B3,
        float* __restrict__ O0, float* __restrict__ O1,
        float* __restrict__ O2, float* __restrict__ O3) {
    int lane  = threadIdx.x;                       // 0..31
    int combo = blockIdx.y * 8 + threadIdx.y;      // 0..15
    int mat   = combo >> 2;
    int ntile = combo & 3;
    int tile  = blockIdx.x;                        // 6250 node tiles of 16 rows

    const float* W; const float* B; float* O;
    switch (mat) {
        case 0:  W = W0; B = B0; O = O0; break;
        case 1:  W = W1; B = B1; O = O1; break;
        case 2:  W = W2; B = B2; O = O2; break;
        default: W = W3; B = B3; O = O3; break;
    }

    int ncol  = ntile * 16 + (lane & 15);
    int khalf = (lane >> 4) * 2;

    // Preload all 16 B fragments (weights; L2-resident, reused by every node tile)
    v2f bf[16];
#pragma unroll
    for (int s = 0; s < 16; s++) {
        int kb = 4 * s + khalf;
        bf[s].x = W[ncol * 64 + kb];
        bf[s].y = W[ncol * 64 + kb + 1];
    }

    const float* arow = in + (size_t)(tile * 16 + (lane & 15)) * 64;
    v8f acc = {};
#pragma unroll
    for (int s = 0; s < 16; s++) {
        int kb = 4 * s + khalf;
        v2f a;
        a.x = arow[kb];
        a.y = arow[kb + 1];
        acc = __builtin_amdgcn_wmma_f32_16x16x4_f32(false, a, false, bf[s],
                                                    (short)0, acc, false, false);
    }

    float bias = B[ntile * 16 + (lane & 15)];
    int mbase  = tile * 16 + ((lane >> 4) << 3);
    int col    = ntile * 16 + (lane & 15);
#pragma unroll
    for (int i = 0; i < 8; i++)
        O[(size_t)(mbase + i) * 64 + col] = acc[i] + bias;
}

// Edge phase: 32 lanes per edge, float2 per lane (coalesced 256B row reads),
// gate = sigmoid(k[dst]+q[src]); scatter gate*v[src] into agg[dst] with f32 atomics.
__global__ void edge_kernel(const float* __restrict__ K, const float* __restrict__ Q,
                            const float* __restrict__ V,
                            const int* __restrict__ src, const int* __restrict__ dst,
                            float* __restrict__ AGG) {
    int gid = blockIdx.x * blockDim.x + threadIdx.x;
    int e = gid >> 5;
    if (e >= N_EDGES) return;
    int h2 = gid & 31;
    int s = src[e], d = dst[e];
    const float2* k2 = (const float2*)K;
    const float2* q2 = (const float2*)Q;
    const float2* v2 = (const float2*)V;
    float2 kv = k2[(size_t)d * 32 + h2];
    float2 qv = q2[(size_t)s * 32 + h2];
    float2 vv = v2[(size_t)s * 32 + h2];
    float g0 = sigmoidf(kv.x + qv.x);
    float g1 = sigmoidf(kv.y + qv.y);
    atomicAdd(&AGG[(size_t)d * 64 + 2 * h2],     g0 * vv.x);
    atomicAdd(&AGG[(size_t)d * 64 + 2 * h2 + 1], g1 * vv.y);
}

// In-place ReLU + per-channel sum / sumsq (grid stride is a multiple of 64,
// so each thread accumulates a fixed channel in registers -> LDS -> global).
__global__ __launch_bounds__(256) void relu_stats_kernel(float* __restrict__ H,
                                                         float* __restrict__ stats) {
    __shared__ float ssum[64], ssq[64];
    int tid = threadIdx.x;
    if (tid < 64) { ssum[tid] = 0.f; ssq[tid] = 0.f; }
    __syncthreads();
    int idx = blockIdx.x * blockDim.x + tid;
    const int stride = 512 * 256;   // multiple of 64
    float ls = 0.f, lq = 0.f;
    for (; idx < N_NODES * HID; idx += stride) {
        float v = H[idx];
        v = v > 0.f ? v : 0.f;
        H[idx] = v;
        ls += v;
        lq += v * v;
    }
    atomicAdd(&ssum[tid & 63], ls);
    atomicAdd(&ssq[tid & 63], lq);
    __syncthreads();
    if (tid < 64) {
        atomicAdd(&stats[tid], ssum[tid]);
        atomicAdd(&stats[64 + tid], ssq[tid]);
    }
}

__global__ void bn_apply_kernel(float* __restrict__ H, const float* __restrict__ stats,
                                const float* __restrict__ gamma, const float* __restrict__ beta) {
    int gid = blockIdx.x * blockDim.x + threadIdx.x;
    if (gid >= N_NODES * HID) return;
    int c = gid & 63;
    const float inv_n = 1.0f / (float)N_NODES;
    float mu  = stats[c] * inv_n;
    float var = stats[64 + c] * inv_n - mu * mu;
    float rs  = rsqrtf(var + EPSV);
    H[gid] = (H[gid] - mu) * rs * gamma[c] + beta[c];
}

// Mean pool: batch is sorted -> stage per-graph sums in LDS, flush nonzero slots.
__global__ __launch_bounds__(256) void pool_kernel(const float* __restrict__ H,
                                                   const int* __restrict__ batch,
                                                   float* __restrict__ gsum,
                                                   float* __restrict__ gcnt) {
    __shared__ float ls[N_GRAPHS * HID];
    __shared__ float lc[N_GRAPHS];
    int tid = threadIdx.x;
    for (int i = tid; i < N_GRAPHS * HID; i += 256) ls[i] = 0.f;
    if (tid < N_GRAPHS) lc[tid] = 0.f;
    __syncthreads();
    int base = blockIdx.x * 256;
    int c = tid & 63;
    int sub = tid >> 6;                 // 4 nodes in flight
    for (int it = 0; it < 64; it++) {
        int node = base + it * 4 + sub;
        if (node < N_NODES) {
            int g = batch[node];
            atomicAdd(&ls[g * 64 + c], H[(size_t)node * 64 + c]);
            if (c == 0) atomicAdd(&lc[g], 1.0f);
        }
    }
    __syncthreads();
    for (int i = tid; i < N_GRAPHS * HID; i += 256) {
        float v = ls[i];
        if (v != 0.f) atomicAdd(&gsum[i], v);
    }
    if (tid < N_GRAPHS && lc[tid] != 0.f) atomicAdd(&gcnt[tid], lc[tid]);
}

// Head: ft -> layernorm -> relu -> gated attention -> regression. One thread per graph.
__global__ __launch_bounds__(64) void head_kernel(const float* __restrict__ gsum,
        const float* __restrict__ gcnt,
        const float* __restrict__ ftW, const float* __restrict__ ftb,
        const float* __restrict__ lng, const float* __restrict__ lnb,
        const float* __restrict__ a1W, const float* __restrict__ a1b,
        const float* __restrict__ a2W, const float* __restrict__ a2b,
        const float* __restrict__ rW,  const float* __restrict__ rb,
        float* __restrict__ out) {
    __shared__ float sG[N_GRAPHS * HID];
    __shared__ float sT[N_GRAPHS * HID];
    int tid = threadIdx.x;    // 0..63 == graph row
    for (int i = tid; i < N_GRAPHS * HID; i += 64) {
        int r = i >> 6;
        float cnt = gcnt[r];
        cnt = cnt > 1.f ? cnt : 1.f;
        sG[i] = gsum[i] / cnt;
    }
    __syncthreads();
    int r = tid;
    for (int c = 0; c < HID; c++) {
        float a = ftb[c];
        for (int k = 0; k < HID; k++) a += sG[r * 64 + k] * ftW[c * 64 + k];
        sT[r * 64 + c] = a;
    }
    float mu = 0.f;
    for (int k = 0; k < HID; k++) mu += sT[r * 64 + k];
    mu *= (1.0f / HID);
    float var = 0.f;
    for (int k = 0; k < HID; k++) { float d = sT[r * 64 + k] - mu; var += d * d; }
    var *= (1.0f / HID);
    float rs = rsqrtf(var + EPSV);
    for (int k = 0; k < HID; k++) {
        float t = (sT[r * 64 + k] - mu) * rs * lng[k] + lnb[k];
        sT[r * 64 + k] = t > 0.f ? t : 0.f;
    }
    float z = a2b[0];
    for (int j = 0; j < 16; j++) {
        float a = a1b[j];
        for (int k = 0; k < HID; k++) a += sT[r * 64 + k] * a1W[j * 64 + k];
        a = a > 0.f ? a : 0.f;
        z += a * a2W[j];
    }
    z = sigmoidf(z);
    float o = rb[0];
    for (int k = 0; k < HID; k++) o += sT[r * 64 + k] * z * rW[k];
    out[r] = o;
}

extern "C" void kernel_launch(void* const* d_in, const int* in_sizes, int n_in,
                              void* d_out, int out_size, void* d_ws, size_t ws_size,
                              hipStream_t stream) {
    (void)in_sizes; (void)n_in; (void)out_size; (void)ws_size;
    // Input map (insertion-order flatten): 0=X; conv li: 1+(li-1)*10 ..
    //   [kW,kb,qW,qb,vW,vb,sW,sb,gamma,beta]; head 51..60; edge_index 61; batch 62.
    const float* X   = (const float*)d_in[0];
    const int* ei    = (const int*)d_in[61];
    const int* batch = (const int*)d_in[62];
    const int* src = ei;
    const int* dst = ei + N_EDGES;

    const size_t S = (size_t)N_NODES * HID;
    float* ws   = (float*)d_ws;
    float* Kb   = ws;
    float* Qb   = ws + S;
    float* Vb   = ws + 2 * S;
    float* bufA = ws + 3 * S;
    float* bufB = ws + 4 * S;
    float* stats = ws + 5 * S;        // 128 floats (sum, sumsq)
    float* gsum  = stats + 128;       // 4096
    float* gcnt  = gsum + 4096;       // 64 (contiguous with gsum)

    float* hin = nullptr;
    float* agg = bufA;
    for (int li = 1; li <= 5; li++) {
        int base = 1 + (li - 1) * 10;
        const float* Wk = (const float*)d_in[base + 0]; const float* bk = (const float*)d_in[base + 1];
        const float* Wq = (const float*)d_in[base + 2]; const float* bq = (const float*)d_in[base + 3];
        const float* Wv = (const float*)d_in[base + 4]; const float* bv = (const float*)d_in[base + 5];
        const float* Wsk = (const float*)d_in[base + 6]; const float* bs = (const float*)d_in[base + 7];
        const float* gam = (const float*)d_in[base + 8]; const float* bet = (const float*)d_in[base + 9];

        zero_kernel<<<1, 128, 0, stream>>>(stats, 128);
        if (li == 1) {
            lin1_kernel<<<25000, 256, 0, stream>>>(X, Wk, bk, Wq, bq, Wv, bv, Wsk, bs,
                                                   Kb, Qb, Vb, agg);
        } else {
            lin_wmma_kernel<<<dim3(6250, 2), dim3(32, 8), 0, stream>>>(hin,
                    Wk, bk, Wq, bq, Wv, bv, Wsk, bs, Kb, Qb, Vb, agg);
        }
        edge_kernel<<<125000, 256, 0, stream>>>(Kb, Qb, Vb, src, dst, agg);
        relu_stats_kernel<<<512, 256, 0, stream>>>(agg, stats);
        bn_apply_kernel<<<25000, 256, 0, stream>>>(agg, stats, gam, bet);

        hin = agg;
        agg = (hin == bufA) ? bufB : bufA;
    }

    zero_kernel<<<(4160 + 127) / 128, 128, 0, stream>>>(gsum, 4160);  // covers gsum + gcnt
    pool_kernel<<<(N_NODES + 255) / 256, 256, 0, stream>>>(hin, batch, gsum, gcnt);
    head_kernel<<<1, 64, 0, stream>>>(gsum, gcnt,
            (const float*)d_in[51], (const float*)d_in[52],
            (const float*)d_in[53], (const float*)d_in[54],
            (const float*)d_in[55], (const float*)d_in[56],
            (const float*)d_in[57], (const float*)d_in[58],
            (const float*)d_in[59], (const float*)d_in[60],
            (float*)d_out);
}